// WeightedStateLossModel4_46995532153319
// MI455X (gfx1250) — compile-verified
//
#include <hip/hip_runtime.h>
#include <math.h>

typedef __attribute__((ext_vector_type(2))) float v2f;
typedef __attribute__((ext_vector_type(8))) float v8f;

#define HDIM 128
#define DDIM 64

// One wave32 per sample; 8 samples per 256-thread block.
__global__ __launch_bounds__(256) void wsl_per_sample(const float* __restrict__ pred,
                                                      const float* __restrict__ targ,
                                                      float* __restrict__ ws) {
    const int lane = threadIdx.x & 31;
    const int i    = blockIdx.x * 8 + (threadIdx.x >> 5);

    const float* tb = targ + (size_t)i * (HDIM * DDIM);
    const float* pb = pred + (size_t)i * (HDIM * DDIM);

    // t = number of nonzero targ[i, :, 1] over H=128 (4 strided loads per lane)
    int cnt = 0;
#pragma unroll
    for (int j = 0; j < 4; ++j) {
        float v = tb[(lane + 32 * j) * DDIM + 1];
        cnt += (v != 0.0f) ? 1 : 0;
    }
#pragma unroll
    for (int off = 16; off > 0; off >>= 1) cnt += __shfl_xor(cnt, off, 32);
    const int t   = cnt;                         // uniform across wave
    const int idx = (t == 0) ? (HDIM - 1) : (t - 1);

    // w = 1 + (K-1) * (t/(H-1))^2.5 ;  x^2.5 = x*x*sqrt(x)
    const float x   = (float)t * (1.0f / (float)(HDIM - 1));
    const float wgt = 1.0f + 0.5f * x * x * sqrtf(x);

    // Row at idx: diff^2 + |diff - prev_diff| ; prev_diff = 0 iff idx == 0.
    // Each lane handles 2 contiguous d-elements -> global_load_b64, coalesced.
    const v2f* pr = (const v2f*)(pb + (size_t)idx * DDIM);
    const v2f* tr = (const v2f*)(tb + (size_t)idx * DDIM);
    v2f dp = pr[lane] - tr[lane];

    v2f prev = {0.0f, 0.0f};
    if (idx > 0) {                                // wave-uniform branch
        const v2f* pp = (const v2f*)(pb + (size_t)(idx - 1) * DDIM);
        const v2f* tp = (const v2f*)(tb + (size_t)(idx - 1) * DDIM);
        prev = pp[lane] - tp[lane];
    }

    float s = dp.x * dp.x + fabsf(dp.x - prev.x)
            + dp.y * dp.y + fabsf(dp.y - prev.y);
#pragma unroll
    for (int off = 16; off > 0; off >>= 1) s += __shfl_xor(s, off, 32);

    if (lane == 0) ws[i] = wgt * s;
}

// Deterministic final reduction of B=4096 partials using the CDNA5 tensor path:
// view ws as A[16][256] (f32) and accumulate C += A_tile(16x4) x ones(4x16)
// with V_WMMA_F32_16X16X4_F32. Every column of C holds the per-row sums, so
// lane 0 (rows 0-7 in v0..v7) + lane 16 (rows 8-15) give the grand total.
__global__ __launch_bounds__(32) void wsl_reduce_wmma(const float* __restrict__ ws,
                                                      float* __restrict__ out, int B) {
    const int lane  = threadIdx.x;          // single wave32, EXEC all ones
    const int m     = lane & 15;            // A-matrix row for this lane
    const int khalf = (lane >> 4) << 1;     // lanes 0-15 -> K=0,1 ; lanes 16-31 -> K=2,3
    const int Ktot  = B >> 4;               // 256 columns

    v8f c = {};
    v2f bones = {1.0f, 1.0f};               // ones matrix: layout-independent

    for (int k0 = 0; k0 < Ktot; k0 += 4) {
        const float* p = ws + (size_t)m * Ktot + k0 + khalf;
        v2f a;
        a.x = p[0];
        a.y = p[1];
        c = __builtin_amdgcn_wmma_f32_16x16x4_f32(
                /*neg_a=*/false, a, /*neg_b=*/false, bones,
                /*c_mod=*/(short)0, c, /*reuse_a=*/false, /*reuse_b=*/false);
    }

    float cs = c[0] + c[1] + c[2] + c[3] + c[4] + c[5] + c[6] + c[7];
    cs += __shfl_xor(cs, 16, 32);           // rows 0-7 + rows 8-15
    if (lane == 0) out[0] = cs / (float)B;  // mean over samples
}

extern "C" void kernel_launch(void* const* d_in, const int* in_sizes, int n_in,
                              void* d_out, int out_size, void* d_ws, size_t ws_size,
                              hipStream_t stream) {
    const float* pred = (const float*)d_in[0];
    const float* targ = (const float*)d_in[1];
    const int B = in_sizes[0] / (HDIM * DDIM);   // 4096

    float* ws = (float*)d_ws;                    // needs B*4 = 16 KB

    wsl_per_sample<<<B / 8, 256, 0, stream>>>(pred, targ, ws);
    wsl_reduce_wmma<<<1, 32, 0, stream>>>(ws, (float*)d_out, B);
}